// Encoder_70471823393245
// MI455X (gfx1250) — compile-verified
//
#include <hip/hip_runtime.h>

typedef float v2f __attribute__((ext_vector_type(2)));
typedef float v8f __attribute__((ext_vector_type(8)));

#if defined(__HIP_DEVICE_COMPILE__) && !__has_builtin(__builtin_amdgcn_wmma_f32_16x16x4_f32)
#error "missing __builtin_amdgcn_wmma_f32_16x16x4_f32 on this toolchain (device pass)"
#endif

// ---------------- degree kernels ----------------
__global__ void k_init_deg(float* __restrict__ deg, int n) {
    int i = blockIdx.x * blockDim.x + threadIdx.x;
    if (i < n) deg[i] = 1.0f;  // self-loop contribution
}

__global__ void k_count_deg(const int* __restrict__ dst, float* __restrict__ deg, int e) {
    int i = blockIdx.x * blockDim.x + threadIdx.x;
    if (i < e) atomicAdd(&deg[dst[i]], 1.0f);
}

__global__ void k_rsqrt(float* __restrict__ deg, int n) {
    int i = blockIdx.x * blockDim.x + threadIdx.x;
    if (i < n) deg[i] = rsqrtf(deg[i]);  // deg -> dis in place
}

// ---------------- GEMM: C[M,Ncols] = A[M,K] @ B[K,Ncols] via V_WMMA_F32_16X16X4_F32
// One wave computes one 16x16 output tile, stepping K by 4.
// A layout  (16x4 f32): lanes 0-15 row M=l hold K=kk+0,kk+1 ; lanes 16-31 hold K=kk+2,kk+3
// B layout  (4x16 f32): lanes 0-15 col N=l hold K rows kk+0,kk+1 ; lanes 16-31 rows kk+2,kk+3
// C/D layout: VGPR r -> row r (lanes 0-15) / row r+8 (lanes 16-31), col = lane&15
__global__ void k_gemm_wmma_f32(const float* __restrict__ A, const float* __restrict__ Bm,
                                float* __restrict__ C, int M, int K, int Ncols) {
#if defined(__HIP_DEVICE_COMPILE__)
    int tiles_n = Ncols >> 4;
    int tiles_m = (M + 15) >> 4;
    int wave = blockIdx.x * (blockDim.x >> 5) + (threadIdx.x >> 5);
    if (wave >= tiles_m * tiles_n) return;  // whole-wave exit: EXEC stays all-ones for WMMA
    int lane = threadIdx.x & 31;
    int tm = wave / tiles_n;
    int tn = wave - tm * tiles_n;
    int row0 = tm << 4, col0 = tn << 4;
    int half = lane >> 4;      // 0: lanes 0-15, 1: lanes 16-31
    int l    = lane & 15;
    int arow = row0 + l;
    if (arow >= M) arow = M - 1;  // clamp loads (keeps EXEC full); stores predicated below

    v8f acc = {0.f, 0.f, 0.f, 0.f, 0.f, 0.f, 0.f, 0.f};
    for (int kk = 0; kk < K; kk += 4) {
        int ka = kk + half * 2;
        v2f a, b;
        a[0] = A[(size_t)arow * K + ka];
        a[1] = A[(size_t)arow * K + ka + 1];
        b[0] = Bm[(size_t)(ka + 0) * Ncols + col0 + l];
        b[1] = Bm[(size_t)(ka + 1) * Ncols + col0 + l];
        acc = __builtin_amdgcn_wmma_f32_16x16x4_f32(false, a, false, b, (short)0, acc,
                                                    false, false);
    }
    int rbase = row0 + half * 8;
#pragma unroll
    for (int r = 0; r < 8; ++r) {
        int rr = rbase + r;
        if (rr < M) C[(size_t)rr * Ncols + col0 + l] = acc[r];
    }
#endif
}

// ---------------- self-loop init: agg[i,f] = dis[i]^2 * h[i,f] (also initializes accumulator; no zero-fill needed)
__global__ void k_self_init(const float* __restrict__ dis, const float* __restrict__ h,
                            float* __restrict__ agg, long total, int logF) {
    long i = (long)blockIdx.x * blockDim.x + threadIdx.x;
    if (i >= total) return;
    int node = (int)(i >> logF);
    float s = dis[node];
    agg[i] = s * s * h[i];
}

// ---------------- edge scatter: agg[dst] += dis[src]*dis[dst] * h[src]
// One wave handles EPW edges: lanes 0..EPW-1 load indices coalesced, broadcast via __shfl,
// all 32 lanes gather the feature row coalesced and atomic-add into agg.
template <int F, int EPW>
__global__ void k_edge_scatter(const int* __restrict__ src, const int* __restrict__ dst,
                               const float* __restrict__ dis, const float* __restrict__ h,
                               float* __restrict__ agg, int nE) {
    int wid = blockIdx.x * (blockDim.x >> 5) + (threadIdx.x >> 5);
    int lane = threadIdx.x & 31;
    long e0 = (long)wid * EPW;
    if (e0 >= nE) return;

    int sv = 0, dv = 0;
    float nv = 0.f;
    if (lane < EPW && (e0 + lane) < nE) {
        sv = src[e0 + lane];
        dv = dst[e0 + lane];
        nv = dis[sv] * dis[dv];
        __builtin_prefetch(&h[(size_t)sv * F], 0, 1);  // global_prefetch_b8 on src rows
    }
    long rem = (long)nE - e0;
    int cnt = rem < EPW ? (int)rem : EPW;
    for (int j = 0; j < cnt; ++j) {
        int s = __shfl(sv, j);
        int d = __shfl(dv, j);
        float nrm = __shfl(nv, j);
#pragma unroll
        for (int f = lane; f < F; f += 32) {
            float v = h[(size_t)s * F + f] * nrm;
            atomicAdd(&agg[(size_t)d * F + f], v);  // global_atomic_add_f32 (no return)
        }
    }
}

// ---------------- bias + relu ----------------
__global__ void k_bias_relu(const float* __restrict__ agg, const float* __restrict__ b,
                            float* __restrict__ out, long total, int fmask) {
    long i = (long)blockIdx.x * blockDim.x + threadIdx.x;
    if (i >= total) return;
    float v = agg[i] + b[(int)i & fmask];
    out[i] = v > 0.f ? v : 0.f;
}

static inline int ceil_div(long a, long b) { return (int)((a + b - 1) / b); }

extern "C" void kernel_launch(void* const* d_in, const int* in_sizes, int n_in,
                              void* d_out, int out_size, void* d_ws, size_t ws_size,
                              hipStream_t stream) {
    const float* x  = (const float*)d_in[0];
    const int*   ei = (const int*)d_in[1];
    const float* b1 = (const float*)d_in[3];
    const float* W1 = (const float*)d_in[2];
    const float* W2 = (const float*)d_in[4];
    const float* b2 = (const float*)d_in[5];
    float* out = (float*)d_out;

    const int IN = 64, HID = 64, OUT = 32;
    int N = in_sizes[0] / IN;
    int E = in_sizes[1] / 2;
    const int* src = ei;      // edge_index[0]
    const int* dst = ei + E;  // edge_index[1]

    // workspace layout (floats): [dis: N][hbuf: N*64][abuf: N*64]
    float* ws   = (float*)d_ws;
    float* dis  = ws;                       // deg -> dis in place
    float* hbuf = ws + N;                   // h (layer1 gemm), then h1 (post-relu)
    float* abuf = hbuf + (size_t)N * HID;   // agg1; later h2 = abuf[0:N*32], agg2 = abuf[N*32:]
    float* h2   = abuf;
    float* agg2 = abuf + (size_t)N * OUT;

    const int TB = 256;

    // --- degrees / normalization ---
    k_init_deg<<<ceil_div(N, TB), TB, 0, stream>>>(dis, N);
    k_count_deg<<<ceil_div(E, TB), TB, 0, stream>>>(dst, dis, E);
    k_rsqrt<<<ceil_div(N, TB), TB, 0, stream>>>(dis, N);

    // --- layer 1: h = X @ W1 ---
    {
        long tiles = (long)((N + 15) / 16) * (HID / 16);
        k_gemm_wmma_f32<<<ceil_div(tiles, TB / 32), TB, 0, stream>>>(x, W1, hbuf, N, IN, HID);
    }
    k_self_init<<<ceil_div((long)N * HID, TB), TB, 0, stream>>>(dis, hbuf, abuf,
                                                               (long)N * HID, 6);
    {
        int groups = ceil_div(E, 8);
        k_edge_scatter<64, 8><<<ceil_div(groups, TB / 32), TB, 0, stream>>>(src, dst, dis,
                                                                            hbuf, abuf, E);
    }
    k_bias_relu<<<ceil_div((long)N * HID, TB), TB, 0, stream>>>(abuf, b1, hbuf,
                                                               (long)N * HID, HID - 1);

    // --- layer 2: h2 = h1 @ W2 ---
    {
        long tiles = (long)((N + 15) / 16) * (OUT / 16);
        k_gemm_wmma_f32<<<ceil_div(tiles, TB / 32), TB, 0, stream>>>(hbuf, W2, h2, N, HID, OUT);
    }
    k_self_init<<<ceil_div((long)N * OUT, TB), TB, 0, stream>>>(dis, h2, agg2,
                                                               (long)N * OUT, 5);
    {
        int groups = ceil_div(E, 8);
        k_edge_scatter<32, 8><<<ceil_div(groups, TB / 32), TB, 0, stream>>>(src, dst, dis,
                                                                            h2, agg2, E);
    }
    k_bias_relu<<<ceil_div((long)N * OUT, TB), TB, 0, stream>>>(agg2, b2, out,
                                                               (long)N * OUT, OUT - 1);
}